// ComplexAttnBlock_61890478735543
// MI455X (gfx1250) — compile-verified
//
#include <hip/hip_runtime.h>
#include <hip/hip_bf16.h>

// Problem constants (match reference)
#define Bn 4
#define Cc 256
#define Hh 48
#define Wn 48
#define Np (Hh * Wn)        // 2304 pixels
#define Gg 32               // GroupNorm groups over 2*C interleaved channels
#define GN_EPS 1e-6f
#define SM_EPS 1e-8f

typedef __attribute__((ext_vector_type(2))) float v2f;
typedef __attribute__((ext_vector_type(8))) float v8f;

// fp32 WMMA 16x16x4: D = A(16x4) * B(4x16) + C(16x16), exact fp32 matrix math.
__device__ __forceinline__ v8f wmma4(v2f a, v2f b, v8f c) {
  return __builtin_amdgcn_wmma_f32_16x16x4_f32(
      /*neg_a=*/false, a, /*neg_b=*/false, b,
      /*c_mod=*/(short)0, c, /*reuse_a=*/false, /*reuse_b=*/false);
}

// Complex GEMM, one wave computes a 16(M) x 16*NT(N) strip of D = A*B.
// A fragments are loaded once per K-step and reused across the NT tiles,
// raising the WMMA:VMEM ratio in the inner loop (~16 WMMA : 20 loads at NT=4).
// ATR: A stored as mem[k*lda + m] (transposed); else mem[m*lda + k].
// BTR: B stored as mem[n*ldb + k] (transposed); else mem[k*ldb + n].
// fp32 A-frag (16x4): lanes 0-15 rows M=lane, VGPR j -> K = 2*hi + j
// fp32 B-frag (4x16): lanes 0-15 cols N=lane, VGPR j -> K = 2*hi + j
// C/D layout: acc[v] -> D[m0 + v + 8*hi, n + (lane&15)]
template <bool ATR, bool BTR, int NT>
__device__ __forceinline__ void cgemm_strip(
    const float* __restrict__ Are, const float* __restrict__ Aim, int lda,
    const float* __restrict__ Bre, const float* __restrict__ Bim, int ldb,
    int m0, int n0, int K, int lane, v8f (&dre)[NT], v8f (&dim)[NT]) {
  const int r = lane & 15;
  const int hi = lane >> 4;
  const v8f zero = {};
  v8f rr[NT], mm[NT], ii[NT];
#pragma unroll
  for (int tt = 0; tt < NT; ++tt) { rr[tt] = zero; mm[tt] = zero; ii[tt] = zero; }

  for (int k0 = 0; k0 < K; k0 += 4) {
    const int ka = k0 + 2 * hi;
    v2f are_f, aim_f;
    if (ATR) {
      are_f[0] = Are[(size_t)(ka + 0) * lda + m0 + r];
      are_f[1] = Are[(size_t)(ka + 1) * lda + m0 + r];
      aim_f[0] = Aim[(size_t)(ka + 0) * lda + m0 + r];
      aim_f[1] = Aim[(size_t)(ka + 1) * lda + m0 + r];
    } else {
      are_f[0] = Are[(size_t)(m0 + r) * lda + ka + 0];
      are_f[1] = Are[(size_t)(m0 + r) * lda + ka + 1];
      aim_f[0] = Aim[(size_t)(m0 + r) * lda + ka + 0];
      aim_f[1] = Aim[(size_t)(m0 + r) * lda + ka + 1];
    }
#pragma unroll
    for (int tt = 0; tt < NT; ++tt) {
      const int nc = n0 + tt * 16 + r;
      v2f bre_f, bim_f;
      if (BTR) {
        bre_f[0] = Bre[(size_t)nc * ldb + ka + 0];
        bre_f[1] = Bre[(size_t)nc * ldb + ka + 1];
        bim_f[0] = Bim[(size_t)nc * ldb + ka + 0];
        bim_f[1] = Bim[(size_t)nc * ldb + ka + 1];
      } else {
        bre_f[0] = Bre[(size_t)(ka + 0) * ldb + nc];
        bre_f[1] = Bre[(size_t)(ka + 1) * ldb + nc];
        bim_f[0] = Bim[(size_t)(ka + 0) * ldb + nc];
        bim_f[1] = Bim[(size_t)(ka + 1) * ldb + nc];
      }
      rr[tt] = wmma4(are_f, bre_f, rr[tt]);
      mm[tt] = wmma4(aim_f, bim_f, mm[tt]);
      ii[tt] = wmma4(are_f, bim_f, ii[tt]);
      ii[tt] = wmma4(aim_f, bre_f, ii[tt]);
    }
  }
#pragma unroll
  for (int tt = 0; tt < NT; ++tt) {
    dre[tt] = rr[tt] - mm[tt];
    dim[tt] = ii[tt];
  }
}

// Block = 8 waves arranged 2(M) x 4(N): block tile 32(M) x 256(N).
// Wave strip: 16(M) x 64(N), NT = 4.
#define NT4 4
__device__ __forceinline__ void strip_coords(int& m0, int& n0, int& lane) {
  lane = threadIdx.x & 31;
  const int wid = threadIdx.x >> 5;
  m0 = blockIdx.y * 32 + (wid >> 2) * 16;
  n0 = blockIdx.x * 256 + (wid & 3) * 64;
}

// ---------------------------------------------------------------------------
// Kernel 1: Complex GroupNorm.  One block per (b, group); group = 8 complex
// channels, stats over 8*Np*2 = 36864 values.
// ---------------------------------------------------------------------------
__global__ void __launch_bounds__(256) gn_kernel(
    const float* __restrict__ x,      // [B, C, Np, 2]
    const float* __restrict__ gamma,  // [2C]
    const float* __restrict__ beta,   // [2C]
    float* __restrict__ hre, float* __restrict__ him)  // [B, C, Np]
{
  const int b = blockIdx.x / Gg;
  const int g = blockIdx.x % Gg;
  const int t = threadIdx.x;
  const int CH = (2 * Cc) / Gg / 2;  // 8 complex channels per group
  const int M = CH * Np * 2;         // 36864 scalar values

  __shared__ float s_sum[256];
  __shared__ float s_sq[256];

  float sum = 0.f, sq = 0.f;
  for (int i = t; i < M; i += 256) {
    const int cl = i / (Np * 2);
    const int rem = i - cl * (Np * 2);
    const int n = rem >> 1;
    const int comp = rem & 1;
    const int c = g * CH + cl;
    const float v = x[(((size_t)b * Cc + c) * Np + n) * 2 + comp];
    sum += v;
    sq += v * v;
  }
  s_sum[t] = sum;
  s_sq[t] = sq;
  __syncthreads();
  for (int s = 128; s > 0; s >>= 1) {
    if (t < s) { s_sum[t] += s_sum[t + s]; s_sq[t] += s_sq[t + s]; }
    __syncthreads();
  }
  const float inv_m = 1.0f / (float)M;
  const float mu = s_sum[0] * inv_m;
  float var = s_sq[0] * inv_m - mu * mu;
  var = fmaxf(var, 0.0f);
  const float rs = rsqrtf(var + GN_EPS);

  for (int i = t; i < M; i += 256) {
    const int cl = i / (Np * 2);
    const int rem = i - cl * (Np * 2);
    const int n = rem >> 1;
    const int comp = rem & 1;
    const int c = g * CH + cl;
    const float v = x[(((size_t)b * Cc + c) * Np + n) * 2 + comp];
    const int qidx = 2 * c + comp;
    const float xn = (v - mu) * rs * gamma[qidx] + beta[qidx];
    const size_t o = ((size_t)b * Cc + c) * Np + n;
    if (comp == 0) hre[o] = xn; else him[o] = xn;
  }
}

// ---------------------------------------------------------------------------
// Kernel 2: complex 1x1 conv  Y = W * H + bias  (W: [2,C,C], H: [B,C,Np])
// Grid: (Np/256, C/32, B), block 256 (8 waves).
// ---------------------------------------------------------------------------
__global__ void __launch_bounds__(256) cgemm_wh_kernel(
    const float* __restrict__ w,     // [2, C, C]
    const float* __restrict__ bias,  // [2, C]
    const float* __restrict__ hre, const float* __restrict__ him,
    float* __restrict__ yre, float* __restrict__ yim) {
  int m0, n0, lane;
  strip_coords(m0, n0, lane);
  const size_t off = (size_t)blockIdx.z * Cc * Np;

  v8f dre[NT4], dim[NT4];
  cgemm_strip<false, false, NT4>(w, w + Cc * Cc, Cc, hre + off, him + off, Np,
                                 m0, n0, Cc, lane, dre, dim);
  const int r = lane & 15;
  const int hi = lane >> 4;
#pragma unroll
  for (int tt = 0; tt < NT4; ++tt) {
#pragma unroll
    for (int v = 0; v < 8; ++v) {
      const int row = m0 + v + 8 * hi;
      const int col = n0 + tt * 16 + r;
      yre[off + (size_t)row * Np + col] = dre[tt][v] + bias[row];
      yim[off + (size_t)row * Np + col] = dim[tt][v] + bias[Cc + row];
    }
  }
}

// ---------------------------------------------------------------------------
// Kernel 3: attention scores  S[nq, mk] = (1/sqrt(C)) * sum_c q[c,nq]*k[c,mk]
// A = Q^T (transposed load, coalesced across lanes), B = K direct.
// Grid: (Np/256 over keys, Np/32 over queries, B)
// ---------------------------------------------------------------------------
__global__ void __launch_bounds__(256) attn_kernel(
    const float* __restrict__ qre, const float* __restrict__ qim,
    const float* __restrict__ kre, const float* __restrict__ kim,
    float* __restrict__ sre, float* __restrict__ sim) {
  int m0, n0, lane;
  strip_coords(m0, n0, lane);
  const size_t off = (size_t)blockIdx.z * Cc * Np;
  const size_t aoff = (size_t)blockIdx.z * Np * Np;

  v8f dre[NT4], dim[NT4];
  cgemm_strip<true, false, NT4>(qre + off, qim + off, Np, kre + off, kim + off,
                                Np, m0, n0, Cc, lane, dre, dim);
  const float scale = 0.0625f;  // C^-0.5, C = 256
  const int r = lane & 15;
  const int hi = lane >> 4;
#pragma unroll
  for (int tt = 0; tt < NT4; ++tt) {
#pragma unroll
    for (int v = 0; v < 8; ++v) {
      const int row = m0 + v + 8 * hi;
      const int col = n0 + tt * 16 + r;
      sre[aoff + (size_t)row * Np + col] = dre[tt][v] * scale;
      sim[aoff + (size_t)row * Np + col] = dim[tt][v] * scale;
    }
  }
}

// ---------------------------------------------------------------------------
// Kernel 4: magnitude softmax with phase preservation, in place on S.
// aw = softmax(|s|) * (cos, sin)(angle(s)) = softmax(|s|) * s/|s|.
// One block per (row, batch); row magnitudes/exps staged in LDS.
// ---------------------------------------------------------------------------
__global__ void __launch_bounds__(256) softmax_kernel(
    float* __restrict__ sre, float* __restrict__ sim) {
  const int row = blockIdx.x;
  const size_t off = (size_t)blockIdx.y * Np * Np + (size_t)row * Np;
  const int t = threadIdx.x;

  __shared__ float s_mag[Np];
  __shared__ float s_e[Np];
  __shared__ float s_red[256];

  float mx = -3.402823466e+38f;
  for (int i = t; i < Np; i += 256) {
    const float re = sre[off + i];
    const float im = sim[off + i];
    const float m = sqrtf(re * re + im * im);
    s_mag[i] = m;
    mx = fmaxf(mx, m);
  }
  s_red[t] = mx;
  __syncthreads();
  for (int s = 128; s > 0; s >>= 1) {
    if (t < s) s_red[t] = fmaxf(s_red[t], s_red[t + s]);
    __syncthreads();
  }
  mx = s_red[0];
  __syncthreads();

  float sum = 0.f;
  for (int i = t; i < Np; i += 256) {
    const float e = expf(s_mag[i] - mx);
    s_e[i] = e;
    sum += e;
  }
  s_red[t] = sum;
  __syncthreads();
  for (int s = 128; s > 0; s >>= 1) {
    if (t < s) s_red[t] += s_red[t + s];
    __syncthreads();
  }
  const float inv_den = 1.0f / (s_red[0] + SM_EPS);

  for (int i = t; i < Np; i += 256) {
    const float re = sre[off + i];
    const float im = sim[off + i];
    const float m = s_mag[i];
    const float sm = s_e[i] * inv_den;
    float c, s;
    if (m > 0.f) {
      const float inv = 1.0f / m;
      c = re * inv;
      s = im * inv;
    } else {  // atan2(0,0)=0 -> cos=1, sin=0
      c = 1.0f;
      s = 0.0f;
    }
    sre[off + i] = sm * c;
    sim[off + i] = sm * s;
  }
}

// ---------------------------------------------------------------------------
// Kernel 5: out[c, n] = sum_m v[c, m] * aw[n, m]   (A = V direct, B = AW^T;
// each lane streams one AW row contiguously over the K loop).
// Grid: (Np/256 over n, C/32 over c, B)
// ---------------------------------------------------------------------------
__global__ void __launch_bounds__(256) av_kernel(
    const float* __restrict__ vre, const float* __restrict__ vim,
    const float* __restrict__ awre, const float* __restrict__ awim,
    float* __restrict__ ore, float* __restrict__ oim) {
  int m0, n0, lane;
  strip_coords(m0, n0, lane);
  const size_t voff = (size_t)blockIdx.z * Cc * Np;
  const size_t aoff = (size_t)blockIdx.z * Np * Np;

  v8f dre[NT4], dim[NT4];
  cgemm_strip<false, true, NT4>(vre + voff, vim + voff, Np, awre + aoff,
                                awim + aoff, Np, m0, n0, Np, lane, dre, dim);
  const int r = lane & 15;
  const int hi = lane >> 4;
#pragma unroll
  for (int tt = 0; tt < NT4; ++tt) {
#pragma unroll
    for (int v = 0; v < 8; ++v) {
      const int row = m0 + v + 8 * hi;
      const int col = n0 + tt * 16 + r;
      ore[voff + (size_t)row * Np + col] = dre[tt][v];
      oim[voff + (size_t)row * Np + col] = dim[tt][v];
    }
  }
}

// ---------------------------------------------------------------------------
// Kernel 6: proj 1x1 conv + bias + residual; writes interleaved complex out.
// ---------------------------------------------------------------------------
__global__ void __launch_bounds__(256) proj_kernel(
    const float* __restrict__ w,     // [2, C, C]
    const float* __restrict__ bias,  // [2, C]
    const float* __restrict__ ore, const float* __restrict__ oim,
    const float* __restrict__ x_ri,  // [B, C, Np, 2]
    float* __restrict__ out)         // [B, C, Np, 2]
{
  int m0, n0, lane;
  strip_coords(m0, n0, lane);
  const size_t off = (size_t)blockIdx.z * Cc * Np;

  v8f dre[NT4], dim[NT4];
  cgemm_strip<false, false, NT4>(w, w + Cc * Cc, Cc, ore + off, oim + off, Np,
                                 m0, n0, Cc, lane, dre, dim);
  const int r = lane & 15;
  const int hi = lane >> 4;
#pragma unroll
  for (int tt = 0; tt < NT4; ++tt) {
#pragma unroll
    for (int v = 0; v < 8; ++v) {
      const int row = m0 + v + 8 * hi;
      const int col = n0 + tt * 16 + r;
      const size_t xi = (off + (size_t)row * Np + col) * 2;
      out[xi + 0] = x_ri[xi + 0] + dre[tt][v] + bias[row];
      out[xi + 1] = x_ri[xi + 1] + dim[tt][v] + bias[Cc + row];
    }
  }
}

// ---------------------------------------------------------------------------
extern "C" void kernel_launch(void* const* d_in, const int* in_sizes, int n_in,
                              void* d_out, int out_size, void* d_ws,
                              size_t ws_size, hipStream_t stream) {
  (void)in_sizes; (void)n_in; (void)out_size; (void)ws_size;

  const float* x_ri = (const float*)d_in[0];
  const float* gn_gamma = (const float*)d_in[1];
  const float* gn_beta = (const float*)d_in[2];
  const float* qw = (const float*)d_in[3];
  const float* qb = (const float*)d_in[4];
  const float* kw = (const float*)d_in[5];
  const float* kb = (const float*)d_in[6];
  const float* vw = (const float*)d_in[7];
  const float* vb = (const float*)d_in[8];
  const float* pw = (const float*)d_in[9];
  const float* pb = (const float*)d_in[10];
  float* out = (float*)d_out;

  // Workspace layout (floats)
  float* ws = (float*)d_ws;
  const size_t PLANE = (size_t)Bn * Cc * Np;   // 2,359,296 floats
  const size_t APLANE = (size_t)Bn * Np * Np;  // 21,233,664 floats
  float* hre = ws + 0 * PLANE;
  float* him = ws + 1 * PLANE;
  float* qre = ws + 2 * PLANE;
  float* qim = ws + 3 * PLANE;
  float* kre = ws + 4 * PLANE;
  float* kim = ws + 5 * PLANE;
  float* vre = ws + 6 * PLANE;
  float* vim = ws + 7 * PLANE;
  float* ore = ws + 8 * PLANE;
  float* oim = ws + 9 * PLANE;
  float* sre = ws + 10 * PLANE;
  float* sim = sre + APLANE;

  // 1) GroupNorm
  gn_kernel<<<dim3(Bn * Gg), dim3(256), 0, stream>>>(x_ri, gn_gamma, gn_beta,
                                                     hre, him);
  // 2) Q, K, V projections  (block tile 32x256)
  const dim3 gproj(Np / 256, Cc / 32, Bn);
  cgemm_wh_kernel<<<gproj, dim3(256), 0, stream>>>(qw, qb, hre, him, qre, qim);
  cgemm_wh_kernel<<<gproj, dim3(256), 0, stream>>>(kw, kb, hre, him, kre, kim);
  cgemm_wh_kernel<<<gproj, dim3(256), 0, stream>>>(vw, vb, hre, him, vre, vim);

  // 3) Attention scores (block tile 32 queries x 256 keys)
  attn_kernel<<<dim3(Np / 256, Np / 32, Bn), dim3(256), 0, stream>>>(
      qre, qim, kre, kim, sre, sim);

  // 4) Magnitude softmax (in place)
  softmax_kernel<<<dim3(Np, Bn), dim3(256), 0, stream>>>(sre, sim);

  // 5) out = V x AW^T
  av_kernel<<<gproj, dim3(256), 0, stream>>>(vre, vim, sre, sim, ore, oim);

  // 6) proj + residual, interleaved complex output
  proj_kernel<<<gproj, dim3(256), 0, stream>>>(pw, pb, ore, oim, x_ri, out);
}